// TsRnnNet_41205916238045
// MI455X (gfx1250) — compile-verified
//
#include <hip/hip_runtime.h>

// ---------------------------------------------------------------------------
// GRU (H=16) forward on MI455X (gfx1250), wave32, WMMA f32 16x16x4.
//
// Orientation: all 16x16 tiles are TRANSPOSED (rows = gate/hidden unit,
// cols = batch). A-matrices are the constant weights (W_ih streamed, W_hh in
// registers) and the recurrent state S = h^T is consumed directly as the WMMA
// B-matrix each step (only a fixed lane-half swap per row is needed).
//
// Layout assumptions (CDNA5 ISA 7.12.2, f32):
//   A 16x4 : lane m (and m+16) = row m;  vgpr0 = {K0 lo | K2 hi}, vgpr1 = {K1 | K3}
//   B 4x16 : lanes = cols;               vgpr0 = {row k0 lo | row k2 hi}, vgpr1 = {k1 | k3}
//   C/D 16x16: vgpr r = row r (lanes 0-15) and row r+8 (lanes 16-31), cols = lane%16
// ---------------------------------------------------------------------------

typedef float v2f __attribute__((ext_vector_type(2)));
typedef float v8f __attribute__((ext_vector_type(8)));

#define HD 16    // hidden size
#define DD 256   // input dim
#define TT 128   // time steps
#define NG 3     // gates (r, z, n)

__device__ __forceinline__ v8f wmma4(v2f a, v2f b, v8f c) {
  // D = A(16x4) * B(4x16) + C   (f32, full precision)
  return __builtin_amdgcn_wmma_f32_16x16x4_f32(
      /*neg_a=*/false, a, /*neg_b=*/false, b,
      /*c_mod=*/(short)0, c, /*reuse_a=*/false, /*reuse_b=*/false);
}

// Fixed lane-half swap (lane ^ 16) via ds_swizzle SWAPX16 (xor=0x10, and=0x1F):
// one DS op, no per-lane address VALU (unlike __shfl_xor).
#if __has_builtin(__builtin_amdgcn_ds_swizzle)
__device__ __forceinline__ float halfswap(float v) {
  return __int_as_float(__builtin_amdgcn_ds_swizzle(__float_as_int(v), 0x401F));
}
#else
__device__ __forceinline__ float halfswap(float v) { return __shfl_xor(v, 16, 32); }
#endif

// gfx1250 has a hardware V_TANH_F32 transcendental; use it if clang exposes it.
#if __has_builtin(__builtin_amdgcn_tanhf)
__device__ __forceinline__ float htanh(float v) { return __builtin_amdgcn_tanhf(v); }
#elif __has_builtin(__builtin_amdgcn_tanh_f32)
__device__ __forceinline__ float htanh(float v) { return __builtin_amdgcn_tanh_f32(v); }
#else
__device__ __forceinline__ float htanh(float v) {
  return 1.0f - 2.0f / (__expf(2.0f * v) + 1.0f);
}
#endif

__device__ __forceinline__ float hsigmoid(float v) {
  // sigmoid(x) = 0.5 + 0.5*tanh(x/2): single trans op instead of exp+rcp
  return fmaf(0.5f, htanh(0.5f * v), 0.5f);
}

__global__ __launch_bounds__(256) void gru_wmma_kernel(
    const float* __restrict__ x,     // (B, 256)
    const float* __restrict__ Wih,   // (48, 256)
    const float* __restrict__ Whh,   // (48, 16)
    const float* __restrict__ bih,   // (48)
    const float* __restrict__ bhh,   // (48)
    const float* __restrict__ fcw,   // (16)
    const float* __restrict__ fcb,   // (1)
    float* __restrict__ out,         // (B, 128) row-major = ys.T
    int B)
{
  const int lane = threadIdx.x & 31;
  const int wave = threadIdx.x >> 5;
  const int tile = blockIdx.x * 8 + wave;        // one wave = 16 batch rows
  if (tile * HD >= B) return;                    // wave-uniform: EXEC stays all-ones

  const long base = (long)tile * HD;             // batch base row
  const int  m    = lane & 15;                   // row-within-tile / batch col
  const int  hl   = lane >> 4;                   // lane half (K-slot selector)
  const bool lo   = (hl == 0);

  // ------------------------------------------------------------------
  // Phase 1: xp^T[g] = W_ih[16g:16g+16,:] @ x_tile^T   (D-layout tiles)
  // ------------------------------------------------------------------
  v8f xp[NG];
  #pragma unroll
  for (int g = 0; g < NG; ++g)
    #pragma unroll
    for (int r = 0; r < 8; ++r) xp[g][r] = 0.0f;

  const float* xrow = x + (base + m) * DD + 2 * hl;      // B-matrix source (x^T)
  const float* w0   = Wih + (16 * 0 + m) * DD + 2 * hl;  // A-matrix sources (W_ih)
  const float* w1   = Wih + (16 * 1 + m) * DD + 2 * hl;
  const float* w2   = Wih + (16 * 2 + m) * DD + 2 * hl;

  // touch the second half of this lane's x row early (global_prefetch_b8)
  __builtin_prefetch(xrow + DD / 2, 0, 1);

  #pragma unroll 4
  for (int k = 0; k < DD; k += 4) {
    v2f bx = *reinterpret_cast<const v2f*>(xrow + k);    // {k+2hl, k+2hl+1}
    v2f a0 = *reinterpret_cast<const v2f*>(w0 + k);
    v2f a1 = *reinterpret_cast<const v2f*>(w1 + k);
    v2f a2 = *reinterpret_cast<const v2f*>(w2 + k);
    xp[0] = wmma4(a0, bx, xp[0]);
    xp[1] = wmma4(a1, bx, xp[1]);
    xp[2] = wmma4(a2, bx, xp[2]);
  }

  // add b_ih (D-layout: row j = r + 8*hl -> bias index 16g + r + 8*hl)
  #pragma unroll
  for (int g = 0; g < NG; ++g)
    #pragma unroll
    for (int r = 0; r < 8; ++r)
      xp[g][r] += bih[16 * g + r + 8 * hl];

  // ------------------------------------------------------------------
  // Phase 2 setup: W_hh in A-layout registers, bias tiles, fc weights
  // ------------------------------------------------------------------
  v2f Ahh[NG][4];
  #pragma unroll
  for (int g = 0; g < NG; ++g)
    #pragma unroll
    for (int c = 0; c < 4; ++c)
      Ahh[g][c] = *reinterpret_cast<const v2f*>(
          Whh + (16 * g + m) * HD + 4 * c + 2 * hl);

  v8f bhhv[NG];
  #pragma unroll
  for (int g = 0; g < NG; ++g)
    #pragma unroll
    for (int r = 0; r < 8; ++r)
      bhhv[g][r] = bhh[16 * g + r + 8 * hl];

  float fw[8];
  #pragma unroll
  for (int r = 0; r < 8; ++r) fw[r] = fcw[r + 8 * hl];
  const float fb = fcb[0];

  v8f S;                                  // S = h^T tile, D-layout; h0 = 0
  #pragma unroll
  for (int r = 0; r < 8; ++r) S[r] = 0.0f;

  float* orow = out + (base + m) * TT;

  // ------------------------------------------------------------------
  // Recurrence: 128 steps, 12 WMMAs/step, everything register-resident
  // ------------------------------------------------------------------
  for (int t4 = 0; t4 < TT / 4; ++t4) {
    float y4[4];
    #pragma unroll
    for (int s = 0; s < 4; ++s) {
      // ---- reshape S (D-layout rows) into B-matrix K-chunk slots ----
      float sw[8];
      #pragma unroll
      for (int r = 0; r < 8; ++r) sw[r] = halfswap(S[r]);
      // chunk c vgpr0 holds rows {4c lo, 4c+2 hi}; vgpr1 holds {4c+1, 4c+3}
      v2f B0 = { lo ? S[0] : sw[2], lo ? S[1] : sw[3] };   // rows 0..3
      v2f B1 = { lo ? S[4] : sw[6], lo ? S[5] : sw[7] };   // rows 4..7
      v2f B2 = { lo ? sw[0] : S[2], lo ? sw[1] : S[3] };   // rows 8..11
      v2f B3 = { lo ? sw[4] : S[6], lo ? sw[5] : S[7] };   // rows 12..15

      // ---- hp^T[g] = W_hh[16g:,:] @ h^T + b_hh tile ----
      v8f hp0 = bhhv[0], hp1 = bhhv[1], hp2 = bhhv[2];
      hp0 = wmma4(Ahh[0][0], B0, hp0);
      hp1 = wmma4(Ahh[1][0], B0, hp1);
      hp2 = wmma4(Ahh[2][0], B0, hp2);
      hp0 = wmma4(Ahh[0][1], B1, hp0);
      hp1 = wmma4(Ahh[1][1], B1, hp1);
      hp2 = wmma4(Ahh[2][1], B1, hp2);
      hp0 = wmma4(Ahh[0][2], B2, hp0);
      hp1 = wmma4(Ahh[1][2], B2, hp1);
      hp2 = wmma4(Ahh[2][2], B2, hp2);
      hp0 = wmma4(Ahh[0][3], B3, hp0);
      hp1 = wmma4(Ahh[1][3], B3, hp1);
      hp2 = wmma4(Ahh[2][3], B3, hp2);

      // ---- gates (elementwise in D-layout: 8 values/lane, 3 trans ops each) ----
      float part = 0.0f;
      #pragma unroll
      for (int r = 0; r < 8; ++r) {
        float rr = hsigmoid(xp[0][r] + hp0[r]);
        float zz = hsigmoid(xp[1][r] + hp1[r]);
        float nn = htanh(fmaf(rr, hp2[r], xp[2][r]));
        float hn = fmaf(zz, S[r] - nn, nn);       // (1-z)*n + z*h
        S[r] = hn;
        part = fmaf(fw[r], hn, part);             // y partial over this half's rows
      }
      // combine the two halves (rows 0-7 + rows 8-15 of the same batch col)
      y4[s] = part + halfswap(part) + fb;
    }
    // coalesced 16B store per batch row every 4 steps
    if (lane < 16) {
      *reinterpret_cast<float4*>(orow + t4 * 4) =
          make_float4(y4[0], y4[1], y4[2], y4[3]);
    }
  }
}

extern "C" void kernel_launch(void* const* d_in, const int* in_sizes, int n_in,
                              void* d_out, int out_size, void* d_ws, size_t ws_size,
                              hipStream_t stream) {
  const float* x   = (const float*)d_in[0];
  const float* Wih = (const float*)d_in[1];
  const float* Whh = (const float*)d_in[2];
  const float* bih = (const float*)d_in[3];
  const float* bhh = (const float*)d_in[4];
  const float* fcw = (const float*)d_in[5];
  const float* fcb = (const float*)d_in[6];
  float* out = (float*)d_out;

  int B = in_sizes[0] / DD;                 // 65536
  int tiles = (B + HD - 1) / HD;            // 4096 waves
  int blocks = (tiles + 7) / 8;             // 8 waves (256 threads) per block
  gru_wmma_kernel<<<blocks, 256, 0, stream>>>(x, Wih, Whh, bih, bhh, fcw, fcb,
                                              out, B);
}